// LinearNonlinearRelease_82841329205926
// MI455X (gfx1250) — compile-verified
//
#include <hip/hip_runtime.h>
#include <hip/hip_bf16.h>

// ---------------- problem constants ----------------
#define DLEN    1048576               // stimulus length
#define NCELLS  14
#define CPAD    16                    // cells padded to 16 for WMMA tiles
#define K0      20                    // front-end kernel taps
#define K1      32                    // iGluSnFR taps
#define STEADY  640                   // 10*FREQ
#define PADDING 690                   // STEADY + (K0-1) + (K1-1)
#define TLEN    (DLEN + 671)          // length of rp / release sequence
#define NTILES  ((TLEN + 15) / 16)    // 65578 WMMA tiles along time
#define TPAD    (NTILES * 16)         // 1049248 (multiple of 32)
#define XPLEN   (DLEN + PADDING)      // padded stimulus length
#define TILES_PER_BLOCK 8             // 8 waves / 256 threads per block

// param block layout (float offsets inside ws)
#define PRM_KERN   0                  // [K0 x CPAD] filter weights, [k*16+c]
#define PRM_SLOPE  320
#define PRM_OFF    336
#define PRM_CP01   352
#define PRM_CP12   368
#define PRM_RRPCAP 384
#define PRM_IPCAP  400
#define PRM_IP0    416
#define PRM_RRP0   432
#define PRM_SCALE  448
#define PRM_BIAS   464
#define PRM_FLOATS 1024               // reserve 4 KB

// ws requirement: 4096 + TPAD*CPAD*4  ~= 67.2 MB (fits in 192MB L2)

typedef float v2f __attribute__((ext_vector_type(2)));
typedef float v8f __attribute__((ext_vector_type(8)));

#if __has_builtin(__builtin_amdgcn_global_load_async_to_lds_b32) && \
    __has_builtin(__builtin_amdgcn_s_wait_asynccnt)
#define HAS_ASYNC_LDS 1
#else
#define HAS_ASYNC_LDS 0
#endif

__device__ __forceinline__ float fast_rcp(float x) {
    return __builtin_amdgcn_rcpf(x);          // v_rcp_f32
}
__device__ __forceinline__ float eluf(float z) {
    return z > 0.0f ? z : (__expf(z) - 1.0f);
}
__device__ __forceinline__ float smooth_clamp_f(float x, float high) {
    float a = eluf(x - 1.0f) + 1.0f;
    float b = eluf(high - 1.0f - a) - high + 1.0f;
    return -b;
}
__device__ __forceinline__ float sigmf(float z) {
    return fast_rcp(1.0f + __expf(-z));
}

// xp[i] = (i < PADDING) ? x[0] : x[i-PADDING], with upper clamp for tile overrun
__device__ __forceinline__ float load_xp(const float* __restrict__ x, int idx) {
    idx = idx > (XPLEN - 1) ? (XPLEN - 1) : idx;
    int xi = idx - PADDING;
    xi = xi < 0 ? 0 : xi;
    return x[xi];
}

#if HAS_ASYNC_LDS
__device__ __forceinline__ void async_copy_b32(const float* g, float* l) {
    __builtin_amdgcn_global_load_async_to_lds_b32(
        (__attribute__((address_space(1))) int*)(g),
        (__attribute__((address_space(3))) int*)(l),
        /*imm offset=*/0, /*cpol=*/0);
}
#endif

// ------------------------------------------------------------------
// Kernel 0: per-cell parameter prep (16 lanes; lanes 14/15 padded)
// ------------------------------------------------------------------
__global__ void __launch_bounds__(32) prep_kernel(
    float* __restrict__ prm,
    const float* __restrict__ lks, const float* __restrict__ soff,
    const float* __restrict__ lsl, const float* __restrict__ lcp01,
    const float* __restrict__ lcp12, const float* __restrict__ lipc,
    const float* __restrict__ lrpc, const float* __restrict__ sip,
    const float* __restrict__ srrp, const float* __restrict__ lfs,
    const float* __restrict__ fb, const float* __restrict__ ct) {
    const int c = threadIdx.x;
    if (c >= CPAD) return;
    const int cs = (c < NCELLS) ? c : 0;
    const float PI = 3.14159265358979323846f;

    const float ks   = expf(lks[cs]);
    const float tau  = 0.05f * ks;                 // tau_r == tau_d
    const float phi  = -PI * (0.2f / 1.4f) * ks;

    float vals[K0];
    float nrm2 = 0.0f;
    #pragma unroll
    for (int k = 0; k < K0; ++k) {
        const float t  = 0.3f - (float)k * (1.0f / 64.0f);
        const float tr = t / tau;
        const float v  = -(tr * tr * tr) / (1.0f + tr)
                         * expf(-tr * tr)
                         * cosf(2.0f * PI * t / phi + 100.0f);
        vals[k] = v;
        nrm2 += v * v;
    }
    const float inv_nrm = rsqrtf(nrm2);
    const float sgn = (c < NCELLS) ? (-ct[cs] * inv_nrm) : 0.0f;  // pad cells -> 0 weights
    #pragma unroll
    for (int k = 0; k < K0; ++k)
        prm[PRM_KERN + k * CPAD + c] = vals[k] * sgn;

    prm[PRM_SLOPE  + c] = expf(lsl[cs]);
    prm[PRM_OFF    + c] = soff[cs];
    prm[PRM_CP01   + c] = expf(lcp01[cs]);
    prm[PRM_CP12   + c] = expf(lcp12[cs]);
    prm[PRM_RRPCAP + c] = expf(smooth_clamp_f(lrpc[cs], 1.0e6f));
    prm[PRM_IPCAP  + c] = expf(smooth_clamp_f(lipc[cs], 1.0e6f));
    prm[PRM_IP0    + c] = sigmf(sip[cs])  * expf(lipc[cs]);
    prm[PRM_RRP0   + c] = sigmf(srrp[cs]) * expf(lrpc[cs]);
    prm[PRM_SCALE  + c] = expf(lfs[cs]);
    prm[PRM_BIAS   + c] = fb[cs];
}

// ------------------------------------------------------------------
// Kernel 1: front-end filter + sigmoid via V_WMMA_F32_16X16X4_F32.
// One wave per tile of 16 timesteps x 16 cells; K0=20 -> 5 WMMAs.
// 8 tiles per 256-thread block (one per wave; EXEC all-1s per wave).
// A (16x4): lanes 0-15 hold K=kb,kb+1; lanes 16-31 hold K=kb+2,kb+3.
// B (4x16): vgpr j, lanes 0-15 -> row kb+j, lanes 16-31 -> row kb+j+2.
// D (16x16): acc[r] = y[t0 + r + hi*8, lane&15].
// ------------------------------------------------------------------
__global__ void __launch_bounds__(32 * TILES_PER_BLOCK) rp_filter_kernel(
    const float* __restrict__ x,
    const float* __restrict__ prm,
    float* __restrict__ buf) {
    const int tile = blockIdx.x * TILES_PER_BLOCK + (threadIdx.x >> 5);
    if (tile >= NTILES) return;                    // uniform per wave
    const int lane = threadIdx.x & 31;
    const int m    = lane & 15;
    const int hi   = lane >> 4;
    const int t0   = tile * 16;

    v8f acc = {0.0f, 0.0f, 0.0f, 0.0f, 0.0f, 0.0f, 0.0f, 0.0f};
    #pragma unroll
    for (int kb = 0; kb < K0; kb += 4) {
        const int ka = kb + (hi ? 2 : 0);
        v2f a, b;
        a.x = load_xp(x, t0 + m + ka);
        a.y = load_xp(x, t0 + m + ka + 1);
        b.x = prm[PRM_KERN + ka * CPAD + m];
        b.y = prm[PRM_KERN + (ka + 1) * CPAD + m];
        acc = __builtin_amdgcn_wmma_f32_16x16x4_f32(
            /*neg_a=*/false, a, /*neg_b=*/false, b,
            /*c_mod=*/(short)0, acc, /*reuse_a=*/false, /*reuse_b=*/false);
    }

    const float sl = prm[PRM_SLOPE + m];
    const float of = prm[PRM_OFF + m];
    #pragma unroll
    for (int r = 0; r < 8; ++r) {
        const int t  = t0 + r + hi * 8;            // t < TPAD by construction
        const float rp = fast_rcp(1.0f + __expf((of - acc[r]) * sl));
        buf[t * CPAD + m] = rp;
    }
}

// ------------------------------------------------------------------
// Kernel 2: the serial pool recurrence. One wave, lane = cell.
// 32-step chunks, double-buffered register prefetch; releases are
// written back IN PLACE over the rp buffer.
// ------------------------------------------------------------------
#define CH 32
__global__ void __launch_bounds__(32) scan_kernel(
    float* __restrict__ buf,
    const float* __restrict__ prm) {
    const int lane = threadIdx.x;
    if (lane >= CPAD) return;

    const float cp01   = prm[PRM_CP01   + lane];
    const float cp12   = prm[PRM_CP12   + lane];
    const float rrpcap = prm[PRM_RRPCAP + lane];
    const float ipcap  = prm[PRM_IPCAP  + lane];
    float ip  = prm[PRM_IP0  + lane];
    float rrp = prm[PRM_RRP0 + lane];

    float cur[CH], nxt[CH];
    #pragma unroll
    for (int j = 0; j < CH; ++j) cur[j] = buf[j * CPAD + lane];

    for (int t0 = 0; t0 < TPAD; t0 += CH) {
        const int tn = t0 + CH;
        if (tn < TPAD) {
            #pragma unroll
            for (int j = 0; j < CH; ++j) nxt[j] = buf[(tn + j) * CPAD + lane];
        }
        if (t0 + 2 * CH < TPAD)
            __builtin_prefetch(&buf[(t0 + 2 * CH) * CPAD + lane], 0, 0);

        #pragma unroll
        for (int j = 0; j < CH; ++j) {
            const float rp       = cur[j];
            const float rel      = rp * rrp;
            const float released = rrp - rel;
            const float rrp_new  = smooth_clamp_f(released + cp12 * ip, rrpcap);
            const float transfer = rrp_new - released;
            ip  = smooth_clamp_f(ip + cp01 - transfer, ipcap);
            rrp = rrp_new;
            cur[j] = rel;
        }
        #pragma unroll
        for (int j = 0; j < CH; ++j) buf[(t0 + j) * CPAD + lane] = cur[j];
        #pragma unroll
        for (int j = 0; j < CH; ++j) cur[j] = nxt[j];
    }
}

// ------------------------------------------------------------------
// Kernel 3: iGluSnFR 32-tap conv + affine, LDS-staged per (cell, 256-t).
// Staging uses GLOBAL_LOAD_ASYNC_TO_LDS_B32 (ASYNCcnt) when available.
// out[c, t] = scale[c] * sum_k tr[t + STEADY + k] * k1[k] + bias[c]
// ------------------------------------------------------------------
__global__ void __launch_bounds__(256) outconv_kernel(
    const float* __restrict__ buf,
    const float* __restrict__ prm,
    const float* __restrict__ k1,
    float* __restrict__ out) {
    __shared__ float s[256 + K1 - 1];
    __shared__ float sk[K1];
    const int c    = blockIdx.y;
    const int t0   = blockIdx.x * 256;
    const int tid  = threadIdx.x;
    const int base = t0 + STEADY;

#if HAS_ASYNC_LDS
    if (tid < K1) async_copy_b32(&k1[tid], &sk[tid]);
    for (int i = tid; i < 256 + K1 - 1; i += 256)
        async_copy_b32(&buf[(base + i) * CPAD + c], &s[i]);
    __builtin_amdgcn_s_wait_asynccnt(0);
    __syncthreads();
#else
    if (tid < K1) sk[tid] = k1[tid];
    for (int i = tid; i < 256 + K1 - 1; i += 256)
        s[i] = buf[(base + i) * CPAD + c];
    __syncthreads();
#endif

    float sum = 0.0f;
    #pragma unroll
    for (int k = 0; k < K1; ++k) sum = fmaf(s[tid + k], sk[k], sum);

    out[c * DLEN + t0 + tid] = sum * prm[PRM_SCALE + c] + prm[PRM_BIAS + c];
}

// ------------------------------------------------------------------
extern "C" void kernel_launch(void* const* d_in, const int* in_sizes, int n_in,
                              void* d_out, int out_size, void* d_ws, size_t ws_size,
                              hipStream_t stream) {
    const float* x     = (const float*)d_in[0];
    const float* lks   = (const float*)d_in[1];
    const float* soff  = (const float*)d_in[2];
    const float* lsl   = (const float*)d_in[3];
    const float* lcp01 = (const float*)d_in[4];
    const float* lcp12 = (const float*)d_in[5];
    const float* lipc  = (const float*)d_in[6];
    const float* lrpc  = (const float*)d_in[7];
    const float* sip   = (const float*)d_in[8];
    const float* srrp  = (const float*)d_in[9];
    const float* lfs   = (const float*)d_in[10];
    const float* fb    = (const float*)d_in[11];
    const float* ct    = (const float*)d_in[12];
    const float* igk   = (const float*)d_in[13];
    float* out = (float*)d_out;

    float* prm = (float*)d_ws;            // 4 KB param block
    float* buf = prm + PRM_FLOATS;        // TPAD*16 floats (rp -> release, in place)

    prep_kernel<<<1, 32, 0, stream>>>(prm, lks, soff, lsl, lcp01, lcp12,
                                      lipc, lrpc, sip, srrp, lfs, fb, ct);
    rp_filter_kernel<<<(NTILES + TILES_PER_BLOCK - 1) / TILES_PER_BLOCK,
                       32 * TILES_PER_BLOCK, 0, stream>>>(x, prm, buf);
    scan_kernel<<<1, 32, 0, stream>>>(buf, prm);
    outconv_kernel<<<dim3(DLEN / 256, NCELLS), 256, 0, stream>>>(buf, prm, igk, out);
}